// CollisionCost_14851996910153
// MI455X (gfx1250) — compile-verified
//
#include <hip/hip_runtime.h>

typedef float v2f __attribute__((ext_vector_type(2)));
typedef float v8f __attribute__((ext_vector_type(8)));

#define RQ_F      5.0f
#define RADIUS2_F 100.0f
#define THRESH_F  4.0f
#define NQ        960      // total query points (exactly 60*16)
#define NTILE_M   3125     // 50000/16 terrain tiles
#define NQT       60
#define TCHUNKS   32       // terrain chunks in grid.x

__global__ void cc_zero_ws(float* ws, int n) {
    int i = blockIdx.x * blockDim.x + threadIdx.x;
    if (i < n) ws[i] = 0.0f;
}

__global__ __launch_bounds__(256) void cc_accum(
    const float* __restrict__ traj,     // (960,3) flattened queries
    const float* __restrict__ terrain,  // (50000,3)
    float* __restrict__ cnt_ws,         // (960)
    float* __restrict__ sum_ws)         // (960)
{
    const int lane = threadIdx.x & 31;
    const int wave = threadIdx.x >> 5;
    const int hi   = lane >> 4;     // 0: K=0,1 half / rows 0-7 ; 1: K=2,3 half / rows 8-15
    const int lc   = lane & 15;
    const int qbase = blockIdx.y * 16;

    // A matrix (16x4): lane<16 -> {x,y} of query lc ; lane>=16 -> {z,0}
    v2f a;
    {
        const float* qp = traj + 3 * (qbase + lc);
        a.x = hi ? qp[2] : qp[0];
        a.y = hi ? 0.0f  : qp[1];
    }

    // loop-invariant per-lane query norms for the 8 C-rows this lane maps to
    float qn2[8];
#pragma unroll
    for (int v = 0; v < 8; ++v) {
        const float* qp = traj + 3 * (qbase + v + 8 * hi);
        float x = qp[0], y = qp[1], z = qp[2];
        qn2[v] = x * x + y * y + z * z;
    }

    float cacc[8], sacc[8];
#pragma unroll
    for (int v = 0; v < 8; ++v) { cacc[v] = 0.0f; sacc[v] = 0.0f; }

    // Wave-uniform tile index, made provably uniform so loop control + tile
    // addressing live in SGPRs (SALU + SADDR-form loads, no per-iter VALU).
    const int slot   = __builtin_amdgcn_readfirstlane(blockIdx.x * 8 + wave);
    const int stride = TCHUNKS * 8;
    const int voff   = 3 * lc;          // loop-invariant per-lane element offset

    for (int tile = slot; tile < NTILE_M; tile += stride) {
        // B matrix (4x16): lane<16 -> {x,y} of terrain point n=lc ; lane>=16 -> {z,0}
        const float* tp = terrain + (size_t)tile * 48 + voff;   // SGPR base + VGPR offset
        float tx = tp[0], ty = tp[1], tz = tp[2];
        float tn2 = tx * tx + ty * ty + tz * tz;    // col-norm, constant per lane
        v2f b;
        b.x = hi ? tz : tx;
        b.y = hi ? 0.0f : ty;

        v8f c = {};
        // (neg_a, A, neg_b, B, c_mod, C, reuse_a, reuse_b)
        c = __builtin_amdgcn_wmma_f32_16x16x4_f32(
                false, a, false, b, (short)0, c, false, false);

#pragma unroll
        for (int v = 0; v < 8; ++v) {
            float d2 = fmaf(-2.0f, c[v], qn2[v] + tn2);
            bool inside = d2 <= RADIUS2_F;                            // d^2 test: no sqrt dep
            // |d2| folds into the v_sqrt_f32 source modifier (free); guards rounding-negative d2
            float d = __builtin_amdgcn_sqrtf(__builtin_fabsf(d2));    // raw v_sqrt_f32 (TRANS)
            cacc[v] += inside ? 1.0f : 0.0f;
            sacc[v] += inside ? d : 0.0f;
        }
    }

    // reduce across the 16 lanes that share each C-row (xor offsets < 16 stay in-half)
#pragma unroll
    for (int v = 0; v < 8; ++v) {
#pragma unroll
        for (int off = 1; off < 16; off <<= 1) {
            cacc[v] += __shfl_xor(cacc[v], off, 32);
            sacc[v] += __shfl_xor(sacc[v], off, 32);
        }
    }
    if (lc == 0) {
#pragma unroll
        for (int v = 0; v < 8; ++v) {
            int q = qbase + v + 8 * hi;
            atomicAdd(cnt_ws + q, cacc[v]);
            atomicAdd(sum_ws + q, sacc[v]);
        }
    }
}

__global__ void cc_finalize(const float* __restrict__ cnt_ws,
                            const float* __restrict__ sum_ws,
                            float* __restrict__ out)
{
    __shared__ float red[64];
    const int g = blockIdx.x;   // 0..23  (b*6+p)
    const int i = threadIdx.x;  // 0..63
    float pp = 0.0f;
    if (i < 40) {
        int q = g * 40 + i;
        float c = cnt_ws[q];
        float s = sum_ws[q];
        float dm = s / fmaxf(c, 1.0f);
        pp = (c > 0.0f) ? fmaf(-(dm * dm), 1.0f / (RQ_F * RQ_F), THRESH_F) : 0.0f;
    }
    red[i] = pp;
    __syncthreads();
    for (int off = 32; off > 0; off >>= 1) {
        if (i < off) red[i] += red[i + off];
        __syncthreads();
    }
    if (i == 0) out[g] = red[0];
}

extern "C" void kernel_launch(void* const* d_in, const int* in_sizes, int n_in,
                              void* d_out, int out_size, void* d_ws, size_t ws_size,
                              hipStream_t stream) {
    const float* traj    = (const float*)d_in[0];   // (4,6,40,3) -> (960,3)
    const float* terrain = (const float*)d_in[1];   // (50000,3)
    float* out    = (float*)d_out;                  // (4,6) = 24 floats
    float* cnt_ws = (float*)d_ws;                   // 960 floats
    float* sum_ws = cnt_ws + NQ;                    // 960 floats

    cc_zero_ws<<<(2 * NQ + 255) / 256, 256, 0, stream>>>(cnt_ws, 2 * NQ);

    dim3 grid(TCHUNKS, NQT);
    cc_accum<<<grid, 256, 0, stream>>>(traj, terrain, cnt_ws, sum_ws);

    cc_finalize<<<24, 64, 0, stream>>>(cnt_ws, sum_ws, out);
}